// DocumentEncoder_1881195675857
// MI455X (gfx1250) — compile-verified
//
#include <hip/hip_runtime.h>
#include <hip/hip_bf16.h>

#define B_DOCS 256
#define L_TOK  512
#define D_DIM  1024

typedef __attribute__((ext_vector_type(2))) float v2f;
typedef __attribute__((ext_vector_type(8))) float v8f;

// ---------------------------------------------------------------------------
// Kernel 1: hidden[b,:] = mean over tokens of emb_table[tokens[b,l]+1, :]
// One workgroup per document. 256 threads, each owns a float4 slice of D.
// Loads are fully coalesced across the 4KB embedding row.
// ---------------------------------------------------------------------------
__global__ __launch_bounds__(256) void doc_hidden_kernel(
    const int* __restrict__ tokens,      // [B, L]
    const float* __restrict__ emb,       // [V+1, D]
    float* __restrict__ hidden)          // [B, D]
{
    const int b   = blockIdx.x;
    const int tid = threadIdx.x;
    const int* tok = tokens + b * L_TOK;

    float4 acc = make_float4(0.f, 0.f, 0.f, 0.f);
    for (int l = 0; l < L_TOK; ++l) {
        const int row = tok[l] + 1;                       // pad -1 -> row 0
        const float4 v = *(const float4*)(emb + (size_t)row * D_DIM + tid * 4);
        acc.x += v.x; acc.y += v.y; acc.z += v.z; acc.w += v.w;
    }
    const float inv = 1.0f / (float)L_TOK;
    float4* hp = (float4*)(hidden + (size_t)b * D_DIM + tid * 4);
    *hp = make_float4(acc.x * inv, acc.y * inv, acc.z * inv, acc.w * inv);
}

// ---------------------------------------------------------------------------
// Kernel 2: u[b,d] = sum_e hidden[b,e] * W_b[d,e]   (i.e. C = hidden * W_b^T)
// FP32 WMMA 16x16x4, wave32, one wave per 16x16 output tile.
//   A (16x4, M=docs):  lanes 0-15 hold M=lane,   K = k0+{0,1} in vgpr {0,1}
//                      lanes 16-31 hold M=lane-16, K = k0+{2,3}
//   B (4x16, N=d_out): lane holds column N; same K split per lane half.
//                      B[k][n] = W_b[n0+n][k]  -> read W_b row n0+n.
//   C (16x16 f32):     8 VGPRs; lanes 0-15 M=0..7, lanes 16-31 M=8..15.
// ---------------------------------------------------------------------------
__global__ __launch_bounds__(32) void u_gemm_wmma_kernel(
    const float* __restrict__ hidden,    // [B, D]
    const float* __restrict__ Wb,        // [D, D] row-major W_b[d][e]
    float* __restrict__ u)               // [B, D]
{
    const int m0   = blockIdx.x * 16;    // doc tile
    const int n0   = blockIdx.y * 16;    // output-feature tile
    const int lane = threadIdx.x;        // 0..31 (wave32)
    const int half = lane >> 4;          // 0: K {0,1} | 1: K {2,3}
    const int lid  = lane & 15;

    const float* __restrict__ Arow = hidden + (size_t)(m0 + lid) * D_DIM;
    const float* __restrict__ Brow = Wb     + (size_t)(n0 + lid) * D_DIM;

    v8f acc = {};
#pragma unroll 4
    for (int k0 = 0; k0 < D_DIM; k0 += 4) {
        const int ka = k0 + (half << 1);
        v2f a, b;
        a.x = Arow[ka + 0];
        a.y = Arow[ka + 1];
        b.x = Brow[ka + 0];
        b.y = Brow[ka + 1];
        // 8 args: (neg_a, A, neg_b, B, c_mod, C, reuse_a, reuse_b)
        acc = __builtin_amdgcn_wmma_f32_16x16x4_f32(
            false, a, false, b, (short)0, acc, false, false);
    }

#pragma unroll
    for (int r = 0; r < 8; ++r) {
        const int m = half * 8 + r;
        u[(size_t)(m0 + m) * D_DIM + (n0 + lid)] = acc[r];
    }
}

// ---------------------------------------------------------------------------
// Kernel 3: per-document scores + softmax + weighted context.
//   scores[l] = emb[b,l,:] . u[b,:]     (8 waves, wave-parallel dot products)
//   attn      = softmax(scores)         (block reduction in LDS)
//   ct[b,:]   = sum_l attn[l]*emb[b,l,:] (coalesced float4 accumulation)
// ---------------------------------------------------------------------------
__global__ __launch_bounds__(256) void doc_attn_kernel(
    const int* __restrict__ tokens,      // [B, L]
    const float* __restrict__ emb,       // [V+1, D]
    const float* __restrict__ u,         // [B, D]
    float* __restrict__ out)             // [B, D]
{
    __shared__ float s_u[D_DIM];         // 4 KB
    __shared__ float s_sc[L_TOK];        // 2 KB
    __shared__ float s_red[256];         // 1 KB

    const int b    = blockIdx.x;
    const int tid  = threadIdx.x;
    const int wave = tid >> 5;
    const int lane = tid & 31;
    const int* tok = tokens + b * L_TOK;

    // stage u[b,:] into LDS (reused by every score dot product)
    for (int i = tid; i < D_DIM; i += 256)
        s_u[i] = u[(size_t)b * D_DIM + i];
    __syncthreads();

    // ---- scores: each wave owns tokens l = wave, wave+8, ... ----
    for (int l = wave; l < L_TOK; l += 8) {
        const int row = tok[l] + 1;
        const float* er = emb + (size_t)row * D_DIM;
        float s = 0.f;
        for (int i = lane; i < D_DIM; i += 32)
            s += er[i] * s_u[i];
        // wave32 shuffle reduction
#pragma unroll
        for (int off = 16; off > 0; off >>= 1)
            s += __shfl_down(s, off, 32);
        if (lane == 0) s_sc[l] = s;
    }
    __syncthreads();

    // ---- softmax over 512 scores (each thread owns 2) ----
    float v0 = s_sc[tid];
    float v1 = s_sc[tid + 256];
    s_red[tid] = fmaxf(v0, v1);
    __syncthreads();
#pragma unroll
    for (int off = 128; off > 0; off >>= 1) {
        if (tid < off) s_red[tid] = fmaxf(s_red[tid], s_red[tid + off]);
        __syncthreads();
    }
    const float gmax = s_red[0];
    __syncthreads();

    const float e0 = __expf(v0 - gmax);
    const float e1 = __expf(v1 - gmax);
    s_red[tid] = e0 + e1;
    __syncthreads();
#pragma unroll
    for (int off = 128; off > 0; off >>= 1) {
        if (tid < off) s_red[tid] += s_red[tid + off];
        __syncthreads();
    }
    const float inv_sum = 1.0f / s_red[0];
    s_sc[tid]       = e0 * inv_sum;
    s_sc[tid + 256] = e1 * inv_sum;
    __syncthreads();

    // ---- context: ct[b, 4t..4t+3] = sum_l attn[l] * emb_row[4t..4t+3] ----
    float4 acc = make_float4(0.f, 0.f, 0.f, 0.f);
    for (int l = 0; l < L_TOK; ++l) {
        const int row = tok[l] + 1;
        const float a = s_sc[l];
        const float4 v = *(const float4*)(emb + (size_t)row * D_DIM + tid * 4);
        acc.x += a * v.x; acc.y += a * v.y;
        acc.z += a * v.z; acc.w += a * v.w;
    }
    *(float4*)(out + (size_t)b * D_DIM + tid * 4) = acc;
}

// ---------------------------------------------------------------------------
extern "C" void kernel_launch(void* const* d_in, const int* in_sizes, int n_in,
                              void* d_out, int out_size, void* d_ws, size_t ws_size,
                              hipStream_t stream) {
    const int*   tokens = (const int*)d_in[0];    // [256, 512] int32
    // d_in[1] = max_len scalar (compile-time constant here)
    const float* emb    = (const float*)d_in[2];  // [50001, 1024] f32
    const float* Wb     = (const float*)d_in[3];  // [1024, 1024] f32
    float*       out    = (float*)d_out;          // [256, 1024] f32

    float* hidden = (float*)d_ws;                        // 1 MB
    float* u      = hidden + (size_t)B_DOCS * D_DIM;     // 1 MB

    // 1) per-doc mean pooling
    doc_hidden_kernel<<<B_DOCS, 256, 0, stream>>>(tokens, emb, hidden);

    // 2) u = hidden * W_b^T via fp32 WMMA (16x16 tiles)
    dim3 ggrid(B_DOCS / 16, D_DIM / 16);
    u_gemm_wmma_kernel<<<ggrid, 32, 0, stream>>>(hidden, Wb, u);

    // 3) scores -> softmax -> context
    doc_attn_kernel<<<B_DOCS, 256, 0, stream>>>(tokens, emb, u, out);
}